// RMSELoss_24240795419564
// MI455X (gfx1250) — compile-verified
//
#include <hip/hip_runtime.h>
#include <math.h>

// ---------------------------------------------------------------------------
// RMSE of per-frame SE(3) error:  sqrt(mean(||R_kf^T R_gt - I||_F^2
//                                        + ||R_kf^T (p_gt - p_kf)||^2)) + 1e-8
// Memory-bound streaming reduction: 96 B/frame, ~100 flop/frame.
// ---------------------------------------------------------------------------

typedef __attribute__((ext_vector_type(2))) float v2f;
typedef __attribute__((ext_vector_type(8))) float v8f;

#define NBLOCKS 2048
#define TPB     256

struct F3 { float x, y, z; };   // forces 12-byte (b96-class) loads

__global__ __launch_bounds__(TPB) void se3_partial_kernel(
    const float* __restrict__ rot_kf, const float* __restrict__ pose_kf,
    const float* __restrict__ rot_gt, const float* __restrict__ pose_gt,
    float* __restrict__ partials, int n)
{
    float s = 0.0f;
    const int stride = gridDim.x * blockDim.x;
    for (int f = blockIdx.x * blockDim.x + threadIdx.x; f < n; f += stride) {
        const F3* Rk = reinterpret_cast<const F3*>(rot_kf + (size_t)9 * f);
        const F3* Rg = reinterpret_cast<const F3*>(rot_gt + (size_t)9 * f);
        const F3  pk = reinterpret_cast<const F3*>(pose_kf)[f];
        const F3  pg = reinterpret_cast<const F3*>(pose_gt)[f];

        F3 k0 = Rk[0], k1 = Rk[1], k2 = Rk[2];   // rows j = 0,1,2 of rot_kf
        F3 g0 = Rg[0], g1 = Rg[1], g2 = Rg[2];   // rows j = 0,1,2 of rot_gt

        const float kf[9] = {k0.x, k0.y, k0.z, k1.x, k1.y, k1.z, k2.x, k2.y, k2.z};
        const float gt[9] = {g0.x, g0.y, g0.z, g1.x, g1.y, g1.z, g2.x, g2.y, g2.z};

        const float dx = pg.x - pk.x;
        const float dy = pg.y - pk.y;
        const float dz = pg.z - pk.z;

        // R_rel[i][k] = sum_j kf[j][i] * gt[j][k]; t_rel[i] = sum_j kf[j][i]*d[j]
#pragma unroll
        for (int i = 0; i < 3; ++i) {
            const float a0 = kf[i], a1 = kf[3 + i], a2 = kf[6 + i];
#pragma unroll
            for (int k = 0; k < 3; ++k) {
                float r = fmaf(a0, gt[k], fmaf(a1, gt[3 + k], a2 * gt[6 + k]));
                r -= (i == k) ? 1.0f : 0.0f;
                s = fmaf(r, r, s);
            }
            const float t = fmaf(a0, dx, fmaf(a1, dy, a2 * dz));
            s = fmaf(t, t, s);
        }
    }

    // wave32 reduction
#pragma unroll
    for (int off = 16; off > 0; off >>= 1) s += __shfl_xor(s, off, 32);

    __shared__ float ls[TPB / 32];
    const int lane = threadIdx.x & 31;
    const int wid  = threadIdx.x >> 5;
    if (lane == 0) ls[wid] = s;
    __syncthreads();
    if (wid == 0) {
        float v = (lane < (TPB / 32)) ? ls[lane] : 0.0f;
#pragma unroll
        for (int off = 4; off > 0; off >>= 1) v += __shfl_xor(v, off, 32);
        if (lane == 0) partials[blockIdx.x] = v;
    }
}

// Single-wave finalize. The last 32-lane reduction is done with
// V_WMMA_F32_16X16X4_F32: lane partials fill the 16x4 A matrix (the HW lane
// layout is a bijection onto the 64 A slots, so any placement sums correctly),
// B = all-ones 4x16  =>  D[m][n] = rowsum_m, and every full D column sums to
// the grand total. EXEC is all ones here (32 threads, no divergence).
__global__ __launch_bounds__(32) void se3_finalize_kernel(
    const float* __restrict__ partials, float* __restrict__ out, int n)
{
    const int lane = threadIdx.x;  // 0..31, uniform trip count (NBLOCKS % 32 == 0)
    float p = 0.0f;
    for (int i = lane; i < NBLOCKS; i += 32) p += partials[i];

    v2f a = {p, 0.0f};
    v2f b = {1.0f, 1.0f};
    v8f c = {};
    v8f d = __builtin_amdgcn_wmma_f32_16x16x4_f32(
        /*neg_a=*/false, a, /*neg_b=*/false, b,
        /*c_mod=*/(short)0, c, /*reuse_a=*/false, /*reuse_b=*/false);

    float s8 = d[0] + d[1] + d[2] + d[3] + d[4] + d[5] + d[6] + d[7];
    const float total = s8 + __shfl_xor(s8, 16, 32);  // fold half-columns

    if (lane == 0) out[0] = sqrtf(total / (float)n) + 1e-8f;
}

extern "C" void kernel_launch(void* const* d_in, const int* in_sizes, int n_in,
                              void* d_out, int out_size, void* d_ws, size_t ws_size,
                              hipStream_t stream)
{
    const float* rot_kf  = (const float*)d_in[0];
    const float* pose_kf = (const float*)d_in[1];
    const float* rot_gt  = (const float*)d_in[2];
    const float* pose_gt = (const float*)d_in[3];
    const int n = in_sizes[0] / 9;

    float* partials = (float*)d_ws;          // NBLOCKS floats (8 KB), fully
                                             // rewritten each call (no init needed)
    float* out = (float*)d_out;

    se3_partial_kernel<<<NBLOCKS, TPB, 0, stream>>>(
        rot_kf, pose_kf, rot_gt, pose_gt, partials, n);
    se3_finalize_kernel<<<1, 32, 0, stream>>>(partials, out, n);
}